// GCN_66288525247270
// MI455X (gfx1250) — compile-verified
//
#include <hip/hip_runtime.h>
#include <hip/hip_bf16.h>

#define Nn   50000
#define Ee   800000
#define INF_ 32
#define Bb_  8
#define Dd   16
#define EPSn 1e-5f

typedef float v2f __attribute__((ext_vector_type(2)));
typedef float v8f __attribute__((ext_vector_type(8)));

__device__ __forceinline__ void atomicAddF(float* p, float v) {
    __hip_atomic_fetch_add(p, v, __ATOMIC_RELAXED, __HIP_MEMORY_SCOPE_AGENT);
}

// ---------------- utility ----------------
__global__ __launch_bounds__(256) void gcn_zerof(float* __restrict__ p, int n) {
    int i = blockIdx.x * blockDim.x + threadIdx.x;
    if (i < n) p[i] = 0.0f;
}

// degrees: outdeg from src, indeg from dst (arrays pre-zeroed)
__global__ __launch_bounds__(256) void gcn_degrees(const int* __restrict__ src,
                                                   const int* __restrict__ dst,
                                                   float* __restrict__ outdeg,
                                                   float* __restrict__ indeg) {
    int e = blockIdx.x * blockDim.x + threadIdx.x;
    if (e < Ee) {
        atomicAddF(&outdeg[src[e]], 1.0f);
        atomicAddF(&indeg[dst[e]], 1.0f);
    }
}

// deg -> clip(deg,1)^-0.5 in place (applied to both arrays as one 2N region)
__global__ __launch_bounds__(256) void gcn_norms(float* __restrict__ p, int n) {
    int i = blockIdx.x * blockDim.x + threadIdx.x;
    if (i < n) p[i] = rsqrtf(fmaxf(p[i], 1.0f));
}

// ---------------- conv1 GEMM: pre[n,d] = (sum_k x[k*N+n]*W1[k,d]) * out_norm[n]
// one wave32 per 16-node tile; 8 chained V_WMMA_F32_16X16X4_F32 (K=32)
__global__ __launch_bounds__(256) void gcn_conv1_wmma(const float* __restrict__ x,   // x + b*IN*N
                                                      const float* __restrict__ W1,  // [32,16] row-major
                                                      const float* __restrict__ onorm,
                                                      float* __restrict__ pre) {     // [N,16]
    const int tile = blockIdx.x * (blockDim.x >> 5) + (threadIdx.x >> 5);
    const int ntiles = Nn / 16;            // 3125
    if (tile >= ntiles) return;            // wave-uniform; EXEC stays all-1s
    const int lane = threadIdx.x & 31;
    const int mrow = lane & 15;
    const int khalf = (lane >> 4) << 1;    // 0 or 2
    const int node0 = tile * 16;

    v8f acc = {};
#pragma unroll
    for (int kk = 0; kk < 8; ++kk) {
        const int k0 = kk * 4 + khalf;
        v2f a, b;
        // A (16x4 f32): lane holds (M=mrow, K=k0) and (M=mrow, K=k0+1)
        a.x = x[k0 * Nn + node0 + mrow];
        a.y = x[(k0 + 1) * Nn + node0 + mrow];
        // B (4x16 f32): lane holds (K=k0, Ncol=mrow) and (K=k0+1, Ncol=mrow)
        b.x = W1[k0 * Dd + mrow];
        b.y = W1[(k0 + 1) * Dd + mrow];
        acc = __builtin_amdgcn_wmma_f32_16x16x4_f32(false, a, false, b,
                                                    (short)0, acc, false, false);
    }
    // D layout: column = mrow; row M = r + 8*(lane>=16)
#pragma unroll
    for (int r = 0; r < 8; ++r) {
        const int node = node0 + r + ((lane >> 4) << 3);
        pre[node * Dd + mrow] = acc[r] * onorm[node];
    }
}

// ---------------- conv2 GEMM: pre[n,d] = (sum_k h[n,k]*W2[k,d]) * out_norm[n]  (K=16)
__global__ __launch_bounds__(256) void gcn_conv2_wmma(const float* __restrict__ h,   // [N,16]
                                                      const float* __restrict__ W2,  // [16,16]
                                                      const float* __restrict__ onorm,
                                                      float* __restrict__ pre) {
    const int tile = blockIdx.x * (blockDim.x >> 5) + (threadIdx.x >> 5);
    const int ntiles = Nn / 16;
    if (tile >= ntiles) return;
    const int lane = threadIdx.x & 31;
    const int mrow = lane & 15;
    const int khalf = (lane >> 4) << 1;
    const int node0 = tile * 16;

    v8f acc = {};
#pragma unroll
    for (int kk = 0; kk < 4; ++kk) {
        const int k0 = kk * 4 + khalf;
        v2f a, b;
        a.x = h[(node0 + mrow) * Dd + k0];
        a.y = h[(node0 + mrow) * Dd + k0 + 1];
        b.x = W2[k0 * Dd + mrow];
        b.y = W2[(k0 + 1) * Dd + mrow];
        acc = __builtin_amdgcn_wmma_f32_16x16x4_f32(false, a, false, b,
                                                    (short)0, acc, false, false);
    }
#pragma unroll
    for (int r = 0; r < 8; ++r) {
        const int node = node0 + r + ((lane >> 4) << 3);
        pre[node * Dd + mrow] = acc[r] * onorm[node];
    }
}

// ---------------- edge scatter: agg[dst,d] += pre[src,d]  (agg pre-zeroed)
__global__ __launch_bounds__(256) void gcn_aggregate(const float* __restrict__ pre,
                                                     const int* __restrict__ src,
                                                     const int* __restrict__ dst,
                                                     float* __restrict__ agg) {
    int idx = blockIdx.x * blockDim.x + threadIdx.x;   // E*16 threads exactly
    if (idx < Ee * Dd) {
        int e = idx >> 4, d = idx & 15;
        atomicAddF(&agg[dst[e] * Dd + d], pre[src[e] * Dd + d]);
    }
}

// ---------------- v = agg*in_norm + bias (in place) + per-channel sum/sumsq
__global__ __launch_bounds__(256) void gcn_finalize_stats(float* __restrict__ v,
                                                          const float* __restrict__ inorm,
                                                          const float* __restrict__ bias,
                                                          float* __restrict__ stats) { // [32] zeroed
    __shared__ float s1[256], s2[256];
    const int d = threadIdx.x & 15;
    const float bd = bias[d];
    float ls = 0.0f, lq = 0.0f;
    const int stride = gridDim.x * blockDim.x;          // multiple of 16 -> d fixed
    for (int idx = blockIdx.x * blockDim.x + threadIdx.x; idx < Nn * Dd; idx += stride) {
        const int n = idx >> 4;
        float val = v[idx] * inorm[n] + bd;
        v[idx] = val;
        ls += val;
        lq += val * val;
    }
    s1[threadIdx.x] = ls; s2[threadIdx.x] = lq;
    __syncthreads();
    if (threadIdx.x < 16) {
        float a = 0.0f, q = 0.0f;
        for (int i = threadIdx.x; i < 256; i += 16) { a += s1[i]; q += s2[i]; }
        atomicAddF(&stats[threadIdx.x], a);
        atomicAddF(&stats[16 + threadIdx.x], q);
    }
}

// ---------------- BN (training-mode, biased var) + ReLU, in place
__global__ __launch_bounds__(256) void gcn_bn_relu(float* __restrict__ v,
                                                   const float* __restrict__ stats,
                                                   const float* __restrict__ gamma,
                                                   const float* __restrict__ beta) {
    int idx = blockIdx.x * blockDim.x + threadIdx.x;
    if (idx < Nn * Dd) {
        const int d = idx & 15;
        const float invN = 1.0f / (float)Nn;
        float mu = stats[d] * invN;
        float var = stats[16 + d] * invN - mu * mu;
        float sc = rsqrtf(var + EPSn) * gamma[d];
        float o = (v[idx] - mu) * sc + beta[d];
        v[idx] = fmaxf(o, 0.0f);
    }
}

// ---------------- node-mean accumulation: hg_b[d] += sum_n h[n,d]
__global__ __launch_bounds__(256) void gcn_node_mean(const float* __restrict__ h,
                                                     float* __restrict__ hg_b) { // &hg[b*16], zeroed
    __shared__ float s1[256];
    float ls = 0.0f;
    const int stride = gridDim.x * blockDim.x;
    for (int idx = blockIdx.x * blockDim.x + threadIdx.x; idx < Nn * Dd; idx += stride)
        ls += h[idx];
    s1[threadIdx.x] = ls;
    __syncthreads();
    if (threadIdx.x < 16) {
        float a = 0.0f;
        for (int i = threadIdx.x; i < 256; i += 16) a += s1[i];
        atomicAddF(&hg_b[threadIdx.x], a);
    }
}

// ---------------- final BatchNorm over the B=8 samples -> out[b,d]
__global__ __launch_bounds__(128) void gcn_final_bn(const float* __restrict__ hg_sums,
                                                    const float* __restrict__ gamma,
                                                    const float* __restrict__ beta,
                                                    float* __restrict__ out) {
    __shared__ float h[Bb_][Dd];
    __shared__ float mu[Dd], inv[Dd];
    const int tid = threadIdx.x;
    const int b = tid >> 4, d = tid & 15;
    h[b][d] = hg_sums[tid] * (1.0f / (float)Nn);
    __syncthreads();
    if (tid < Dd) {
        float m = 0.0f;
        for (int i = 0; i < Bb_; ++i) m += h[i][tid];
        m *= (1.0f / (float)Bb_);
        float v = 0.0f;
        for (int i = 0; i < Bb_; ++i) { float t = h[i][tid] - m; v += t * t; }
        v *= (1.0f / (float)Bb_);
        mu[tid] = m;
        inv[tid] = rsqrtf(v + EPSn);
    }
    __syncthreads();
    out[tid] = (h[b][d] - mu[d]) * inv[d] * gamma[d] + beta[d];
}

extern "C" void kernel_launch(void* const* d_in, const int* in_sizes, int n_in,
                              void* d_out, int out_size, void* d_ws, size_t ws_size,
                              hipStream_t stream) {
    const float* x    = (const float*)d_in[0];
    const float* W1   = (const float*)d_in[1];
    const float* b1   = (const float*)d_in[2];
    const float* g1g  = (const float*)d_in[3];
    const float* g1b  = (const float*)d_in[4];
    const float* W2   = (const float*)d_in[5];
    const float* b2   = (const float*)d_in[6];
    const float* g2g  = (const float*)d_in[7];
    const float* g2b  = (const float*)d_in[8];
    const float* n1g  = (const float*)d_in[9];
    const float* n1b  = (const float*)d_in[10];
    const int*   src  = (const int*)d_in[11];
    const int*   dst  = (const int*)d_in[12];
    float*       out  = (float*)d_out;

    // workspace layout (floats): onorm[N] | inorm[N] | bufA[16N] | bufB[16N] | stats[32] | hg[128]
    float* ws    = (float*)d_ws;
    float* onorm = ws;
    float* inorm = ws + Nn;
    float* bufA  = ws + 2 * Nn;
    float* bufB  = ws + 2 * Nn + 16 * Nn;
    float* stats = ws + 2 * Nn + 32 * Nn;
    float* hg    = stats + 32;

    const int ntiles   = Nn / 16;                      // 3125
    const int gConv    = (ntiles + 7) / 8;             // 8 waves/block
    const int gElem    = (Nn * Dd + 255) / 256;        // 3125
    const int gEdge16  = (Ee * Dd + 255) / 256;        // 50000
    const int gE       = (Ee + 255) / 256;             // 3125
    const int gRed     = 512;                          // grid-stride reductions

    // --- degrees & norms ---
    gcn_zerof<<<(2 * Nn + 255) / 256, 256, 0, stream>>>(onorm, 2 * Nn);
    gcn_zerof<<<1, 256, 0, stream>>>(hg, Bb_ * Dd);
    gcn_degrees<<<gE, 256, 0, stream>>>(src, dst, onorm, inorm);
    gcn_norms<<<(2 * Nn + 255) / 256, 256, 0, stream>>>(onorm, 2 * Nn);

    for (int b = 0; b < Bb_; ++b) {
        const float* xb = x + (size_t)b * INF_ * Nn;
        // ---- layer 1 ----
        gcn_conv1_wmma<<<gConv, 256, 0, stream>>>(xb, W1, onorm, bufA);
        gcn_zerof<<<gElem, 256, 0, stream>>>(bufB, Nn * Dd);
        gcn_zerof<<<1, 32, 0, stream>>>(stats, 32);
        gcn_aggregate<<<gEdge16, 256, 0, stream>>>(bufA, src, dst, bufB);
        gcn_finalize_stats<<<gRed, 256, 0, stream>>>(bufB, inorm, b1, stats);
        gcn_bn_relu<<<gElem, 256, 0, stream>>>(bufB, stats, g1g, g1b);
        // ---- layer 2 ----
        gcn_conv2_wmma<<<gConv, 256, 0, stream>>>(bufB, W2, onorm, bufA);
        gcn_zerof<<<gElem, 256, 0, stream>>>(bufB, Nn * Dd);
        gcn_zerof<<<1, 32, 0, stream>>>(stats, 32);
        gcn_aggregate<<<gEdge16, 256, 0, stream>>>(bufA, src, dst, bufB);
        gcn_finalize_stats<<<gRed, 256, 0, stream>>>(bufB, inorm, b2, stats);
        gcn_bn_relu<<<gElem, 256, 0, stream>>>(bufB, stats, g2g, g2b);
        // ---- readout ----
        gcn_node_mean<<<gRed, 256, 0, stream>>>(bufB, hg + b * Dd);
    }
    // ---- final BN over batch ----
    gcn_final_bn<<<1, 128, 0, stream>>>(hg, n1g, n1b, out);
}